// DirMagGCNConv_52939766890537
// MI455X (gfx1250) — compile-verified
//
#include <hip/hip_runtime.h>
#include <stdint.h>

#define D 128
#define HBITS 21
#define HSIZE (1u << HBITS)
#define HMASK (HSIZE - 1u)
#define HEMPTY 0xFFFFFFFFu

// 2*pi*Q with Q=0.25 -> pi/2 ; ALPHA = 0.5
#define TWO_PI_Q 1.5707963267948966f
#define ALPHA_C 0.5f

typedef float v2f __attribute__((ext_vector_type(2)));
typedef float v8f __attribute__((ext_vector_type(8)));

__device__ __forceinline__ uint32_t hashk(uint32_t k) {
    k *= 2654435761u;
    k ^= k >> 15;
    return k & HMASK;
}

__global__ void fill_u32(uint32_t* __restrict__ p, uint32_t v, int n) {
    int i = blockIdx.x * blockDim.x + threadIdx.x;
    int stride = gridDim.x * blockDim.x;
    for (; i < n; i += stride) p[i] = v;
}

// Wc = ALPHA*W1 + (1-ALPHA)*W2 ; bc = ALPHA*b1 + (1-ALPHA)*b2
__global__ void combine_w(const float* __restrict__ W1, const float* __restrict__ b1,
                          const float* __restrict__ W2, const float* __restrict__ b2,
                          float* __restrict__ Wc, float* __restrict__ bc) {
    int i = blockIdx.x * blockDim.x + threadIdx.x;
    if (i < D * D) Wc[i] = ALPHA_C * W1[i] + (1.0f - ALPHA_C) * W2[i];
    if (i < D)     bc[i] = ALPHA_C * b1[i] + (1.0f - ALPHA_C) * b2[i];
}

__global__ void hash_insert(const int* __restrict__ row, const int* __restrict__ col,
                            int E, int N, uint32_t* __restrict__ table) {
    int e = blockIdx.x * blockDim.x + threadIdx.x;
    if (e >= E) return;
    uint32_t key = (uint32_t)row[e] * (uint32_t)N + (uint32_t)col[e];
    uint32_t s = hashk(key);
    for (;;) {
        uint32_t old = atomicCAS(&table[s], HEMPTY, key);
        if (old == HEMPTY || old == key) break;
        s = (s + 1u) & HMASK;
    }
}

__device__ __forceinline__ bool hash_find(const uint32_t* __restrict__ table, uint32_t key) {
    uint32_t s = hashk(key);
    for (;;) {
        uint32_t v = table[s];
        if (v == key) return true;
        if (v == HEMPTY) return false;
        s = (s + 1u) & HMASK;
    }
}

__global__ void degree_count(const int* __restrict__ row, const int* __restrict__ col,
                             int E, uint32_t* __restrict__ cnt) {
    int e = blockIdx.x * blockDim.x + threadIdx.x;
    if (e >= E) return;
    atomicAdd(&cnt[row[e]], 1u);
    atomicAdd(&cnt[col[e]], 1u);
}

__global__ void make_dinv(const uint32_t* __restrict__ cnt, float* __restrict__ dinv, int N) {
    int u = blockIdx.x * blockDim.x + threadIdx.x;
    if (u >= N) return;
    uint32_t c = cnt[u];
    dinv[u] = c ? rsqrtf(0.5f * (float)c) : 0.0f;
}

// per directed edge: forward-phase coefficients; reverse copy uses (cre, -cim)
__global__ void edge_coef(const int* __restrict__ row, const int* __restrict__ col,
                          int E, int N, const uint32_t* __restrict__ table,
                          const float* __restrict__ dinv,
                          float* __restrict__ cre, float* __restrict__ cim) {
    int e = blockIdx.x * blockDim.x + threadIdx.x;
    if (e >= E) return;
    int r = row[e], c = col[e];
    uint32_t rkey = (uint32_t)c * (uint32_t)N + (uint32_t)r;
    bool hr = hash_find(table, rkey);
    float theta = hr ? 0.0f : TWO_PI_Q;
    float scale = 0.5f * dinv[r] * dinv[c];
    cre[e] = scale * cosf(theta);
    cim[e] = scale * sinf(theta);
}

// one wave (32 lanes) per directed edge; each lane handles 4 of 128 features
__global__ void spmm_scatter(const int* __restrict__ row, const int* __restrict__ col, int E,
                             const float* __restrict__ x,
                             const float* __restrict__ cre, const float* __restrict__ cim,
                             float* __restrict__ yre, float* __restrict__ yim) {
    int t = blockIdx.x * blockDim.x + threadIdx.x;
    int e = t >> 5;
    int lane = t & 31;
    if (e >= E) return;
    int r = row[e], c = col[e];
    float a = cre[e];
    float b = cim[e];
    int f = lane * 4;
    const float4 xc = *(const float4*)(x + (size_t)c * D + f);
    const float4 xr = *(const float4*)(x + (size_t)r * D + f);
    float* yrr = yre + (size_t)r * D + f;   // y_re[row] += a*x[col]
    float* yrc = yre + (size_t)c * D + f;   // y_re[col] += a*x[row]
    float* yir = yim + (size_t)r * D + f;   // y_im[row] += b*x[col]
    float* yic = yim + (size_t)c * D + f;   // y_im[col] += -b*x[row]
    atomicAdd(yrr + 0, a * xc.x); atomicAdd(yrr + 1, a * xc.y);
    atomicAdd(yrr + 2, a * xc.z); atomicAdd(yrr + 3, a * xc.w);
    atomicAdd(yrc + 0, a * xr.x); atomicAdd(yrc + 1, a * xr.y);
    atomicAdd(yrc + 2, a * xr.z); atomicAdd(yrc + 3, a * xr.w);
    atomicAdd(yir + 0, b * xc.x); atomicAdd(yir + 1, b * xc.y);
    atomicAdd(yir + 2, b * xc.z); atomicAdd(yir + 3, b * xc.w);
    atomicAdd(yic + 0, -b * xr.x); atomicAdd(yic + 1, -b * xr.y);
    atomicAdd(yic + 2, -b * xr.z); atomicAdd(yic + 3, -b * xr.w);
}

// one wave per 16x16 output tile: C(16x16) = A(16x128) x Wc(128x16) + bias
// out is [N, 256]: col tiles 0..7 from y_re, 8..15 from y_im (same Wc, bc)
__global__ void gemm_wmma_f32(const float* __restrict__ yre, const float* __restrict__ yim,
                              const float* __restrict__ Wc, const float* __restrict__ bc,
                              float* __restrict__ out, int Mtiles) {
    int wave = (blockIdx.x * blockDim.x + threadIdx.x) >> 5;
    int lane = threadIdx.x & 31;
    int jt = wave & 15;        // output column tile 0..15
    int mt = wave >> 4;        // node row tile
    if (mt >= Mtiles) return;  // wave-uniform guard
    const float* A = (jt < 8) ? yre : yim;
    int wc0 = (jt & 7) * 16;   // column base within Wc
    int m0 = mt * 16;
    int half = lane >> 4;      // 0: K even-pair base 0, 1: base 2
    int l16 = lane & 15;
    const float* arow = A + (size_t)(m0 + l16) * D;

    v8f acc = {};
    #pragma unroll
    for (int k0 = 0; k0 < D; k0 += 4) {
        // A fragment: lane l holds A[m0 + l%16][k0 + 2*(l/16) + {0,1}]
        v2f av = *(const v2f*)(arow + k0 + 2 * half);
        // B fragment: lane l holds Wc[k0 + 2*(l/16) + {0,1}][wc0 + l%16]
        v2f bv;
        bv.x = Wc[(size_t)(k0 + 2 * half) * D + wc0 + l16];
        bv.y = Wc[(size_t)(k0 + 2 * half + 1) * D + wc0 + l16];
        acc = __builtin_amdgcn_wmma_f32_16x16x4_f32(
            false, av, false, bv, (short)0, acc, false, false);
    }

    float bias = bc[wc0 + l16];
    int oc = jt * 16 + l16;    // column in 256-wide output
    #pragma unroll
    for (int i = 0; i < 8; ++i) {
        // C layout: VGPR i, lane l -> row m0 + i + 8*(l/16), col l%16
        out[(size_t)(m0 + i + 8 * half) * 256 + oc] = acc[i] + bias;
    }
}

extern "C" void kernel_launch(void* const* d_in, const int* in_sizes, int n_in,
                              void* d_out, int out_size, void* d_ws, size_t ws_size,
                              hipStream_t stream) {
    const float* x  = (const float*)d_in[0];
    const int*   ei = (const int*)d_in[1];
    const float* W1 = (const float*)d_in[2];
    const float* b1 = (const float*)d_in[3];
    const float* W2 = (const float*)d_in[4];
    const float* b2 = (const float*)d_in[5];
    float* out = (float*)d_out;

    const int N = in_sizes[0] / D;   // 40000
    const int E = in_sizes[1] / 2;   // 640000
    const int* row = ei;
    const int* col = ei + E;

    // workspace carve-up (256B aligned slices)
    char* ws = (char*)d_ws;
    size_t off = 0;
    auto alloc = [&](size_t bytes) -> char* {
        char* p = ws + off;
        off += (bytes + 255) & ~(size_t)255;
        return p;
    };
    uint32_t* table = (uint32_t*)alloc((size_t)HSIZE * 4);
    uint32_t* cnt   = (uint32_t*)alloc((size_t)N * 4);
    float*    dinv  = (float*)alloc((size_t)N * 4);
    float*    cre   = (float*)alloc((size_t)E * 4);
    float*    cim   = (float*)alloc((size_t)E * 4);
    float*    Wc    = (float*)alloc((size_t)D * D * 4);
    float*    bcv   = (float*)alloc((size_t)D * 4);
    float*    yre   = (float*)alloc((size_t)N * D * 4);
    float*    yim   = (float*)alloc((size_t)N * D * 4);

    const int T = 256;
    // 1) clear hash table, degree counts, accumulators
    fill_u32<<<2048, T, 0, stream>>>(table, HEMPTY, (int)HSIZE);
    fill_u32<<<512, T, 0, stream>>>(cnt, 0u, N);
    fill_u32<<<4096, T, 0, stream>>>((uint32_t*)yre, 0u, N * D);
    fill_u32<<<4096, T, 0, stream>>>((uint32_t*)yim, 0u, N * D);
    // 2) combined weights
    combine_w<<<(D * D + T - 1) / T, T, 0, stream>>>(W1, b1, W2, b2, Wc, bcv);
    // 3) reverse-edge hash set
    hash_insert<<<(E + T - 1) / T, T, 0, stream>>>(row, col, E, N, table);
    // 4) symmetrized degrees -> dinv
    degree_count<<<(E + T - 1) / T, T, 0, stream>>>(row, col, E, cnt);
    make_dinv<<<(N + T - 1) / T, T, 0, stream>>>(cnt, dinv, N);
    // 5) per-edge magnetic coefficients
    edge_coef<<<(E + T - 1) / T, T, 0, stream>>>(row, col, E, N, table, dinv, cre, cim);
    // 6) scatter SpMM (wave per edge, 8 edges per block)
    spmm_scatter<<<(E + 7) / 8, T, 0, stream>>>(row, col, E, x, cre, cim, yre, yim);
    // 7) WMMA GEMM epilogue: 16x16 tile per wave, 8 waves per block
    int Mtiles = N / 16;                 // 2500 (N divisible by 16)
    int waves = Mtiles * 16;             // 40000
    gemm_wmma_f32<<<waves / 8, T, 0, stream>>>(yre, yim, Wc, bcv, out, Mtiles);
}